// DynamicTree2_71399536329425
// MI455X (gfx1250) — compile-verified
//
#include <hip/hip_runtime.h>
#include <hip/hip_bf16.h>

// MI455X / gfx1250, wave32. Matrix ops via V_WMMA_F32_16X16X4_F32 (fp32 in,
// fp32 acc) — workload is L2/HBM bound (~105 GFLOP vs 134MB tensors resident
// in 192MB L2), so fp32 WMMA keeps full accuracy at zero cost.
// Weights are staged once per workgroup into LDS (<=256KB of the WGP's 320KB)
// as (2r,2r+1) float2 pairs so each WMMA B-fragment is one ds_load_b64.
// B fragments are batch-loaded per K-step so the ds_load_2addr_b64s issue
// back-to-back and drain with graduated s_wait_dscnt, overlapping LDS latency
// with the WMMA chain.

typedef __attribute__((ext_vector_type(2))) float v2f;
typedef __attribute__((ext_vector_type(8))) float v8f;

__device__ __forceinline__ v8f wmma_f32(v2f a, v2f b, v8f c) {
  // 8 args: (neg_a, A, neg_b, B, c_mod, C, reuse_a, reuse_b)
  return __builtin_amdgcn_wmma_f32_16x16x4_f32(
      false, a, false, b, (short)0, c, false, false);
}

// ---------------------------------------------------------------------------
// Generic GEMM + bias + ReLU:  out[M x Cout] = relu(A[M x R] * W + b)
// DECONV==1: Cout=512, weight w[t][ki][ko]; conv_transpose flips the tap:
//            tap = 3 - col/128, which is *constant per blockIdx.y*, so the
//            staged LDS slab is just a different global base pointer.
// Wave tile: 16 rows x 128 cols (8 accumulators), K stepped by 4.
// ---------------------------------------------------------------------------
template <int DECONV>
__global__ __launch_bounds__(256) void gemm_relu_wmma(
    const float* __restrict__ A, const float* __restrict__ W,
    const float* __restrict__ bias, float* __restrict__ out,
    int M, int R, int Cout) {
  extern __shared__ float ws2[];           // [(R/2)*128] float2-paired weights
  const int cb = blockIdx.y * 128;

  // Global slab for this block's 128 output columns.
  const float* slab;
  int stride;
  if (DECONV) {                            // tap flip, constant per column group
    slab = W + (size_t)(3 - (int)blockIdx.y) * 16384;
    stride = 128;
  } else {
    slab = W + cb;
    stride = Cout;
  }

  // Cooperative staging: pair rows (2r, 2r+1) per column -> ds_load_b64 reads.
  const int P = (R >> 1) * 128;            // number of float2 pairs
  for (int p = threadIdx.x; p < P; p += 256) {
    const int r2 = p >> 7;
    const int c  = p & 127;
    ws2[2 * p + 0] = slab[(size_t)(2 * r2 + 0) * stride + c];
    ws2[2 * p + 1] = slab[(size_t)(2 * r2 + 1) * stride + c];
  }
  __syncthreads();

  const int lane    = threadIdx.x & 31;
  const int wave    = threadIdx.x >> 5;
  const int rowBase = (blockIdx.x * 8 + wave) * 16;
  if (rowBase >= M) return;                // wave-uniform
  const int nIdx  = lane & 15;             // column within 16-wide tile
  const int khalf = (lane >> 4) << 1;      // K sub-offset: 0 or 2

  v8f acc[8];
#pragma unroll
  for (int t = 0; t < 8; ++t) acc[t] = (v8f){0, 0, 0, 0, 0, 0, 0, 0};

  const float* Arow = A + (size_t)(rowBase + nIdx) * R + khalf;

  for (int r = 0; r < R; r += 4) {
    v2f a = *(const v2f*)(Arow + r);       // A frag: K = {r+khalf, r+khalf+1}
    const float* wrow = ws2 + (size_t)((r + khalf) >> 1) * 256;
    v2f b[8];
#pragma unroll
    for (int t = 0; t < 8; ++t)            // issue all ds loads first
      b[t] = *(const v2f*)(wrow + 2 * (t * 16 + nIdx));
#pragma unroll
    for (int t = 0; t < 8; ++t)            // then drain into the WMMA chain
      acc[t] = wmma_f32(a, b[t], acc[t]);
  }

  const int mhi = (lane >> 4) << 3;        // C/D layout: lanes 16-31 -> M+8
#pragma unroll
  for (int t = 0; t < 8; ++t) {
    const int tcol = t * 16 + nIdx;
    const int col  = cb + tcol;
    const float bv = bias[DECONV ? tcol : col];
#pragma unroll
    for (int v = 0; v < 8; ++v) {
      const float val = fmaxf(acc[t][v] + bv, 0.0f);
      out[(size_t)(rowBase + mhi + v) * Cout + col] = val;
    }
  }
}

// ---------------------------------------------------------------------------
// Conv1D k=3 SAME + bias + ReLU, Cout=128. Same WMMA tiling; A fetch is
// row-shifted (tap-1), branchless zero padding via clamp + cndmask.
// Cin % 4 == 0 so each K=4 chunk stays within one tap.
// ---------------------------------------------------------------------------
__global__ __launch_bounds__(256) void conv3_relu_wmma(
    const float* __restrict__ A, const float* __restrict__ W,
    const float* __restrict__ bias, float* __restrict__ out,
    int M, int Cin) {
  extern __shared__ float ws2[];           // [(3*Cin/2)*128] float2 pairs
  const int R = 3 * Cin;
  const int P = (R >> 1) * 128;
  for (int p = threadIdx.x; p < P; p += 256) {
    const int r2 = p >> 7;
    const int c  = p & 127;
    ws2[2 * p + 0] = W[(size_t)(2 * r2 + 0) * 128 + c];
    ws2[2 * p + 1] = W[(size_t)(2 * r2 + 1) * 128 + c];
  }
  __syncthreads();

  const int lane    = threadIdx.x & 31;
  const int wave    = threadIdx.x >> 5;
  const int rowBase = (blockIdx.x * 8 + wave) * 16;
  if (rowBase >= M) return;
  const int nIdx  = lane & 15;
  const int khalf = (lane >> 4) << 1;
  const int row   = rowBase + nIdx;

  v8f acc[8];
#pragma unroll
  for (int t = 0; t < 8; ++t) acc[t] = (v8f){0, 0, 0, 0, 0, 0, 0, 0};

  int tap = 0, ki0 = 0;
  for (int r = 0; r < R; r += 4) {
    const int srcRow = row + tap - 1;                  // SAME pad: taps -1,0,+1
    const bool valid = (unsigned)srcRow < (unsigned)M;
    const int  rsafe = valid ? srcRow : row;           // in-range dummy row
    v2f a = *(const v2f*)(A + (size_t)rsafe * Cin + ki0 + khalf);
    if (!valid) a = (v2f){0.0f, 0.0f};                 // v_cndmask, no branch
    const float* wrow = ws2 + (size_t)((r + khalf) >> 1) * 256;
    v2f b[8];
#pragma unroll
    for (int t = 0; t < 8; ++t)
      b[t] = *(const v2f*)(wrow + 2 * (t * 16 + nIdx));
#pragma unroll
    for (int t = 0; t < 8; ++t)
      acc[t] = wmma_f32(a, b[t], acc[t]);
    ki0 += 4;
    if (ki0 == Cin) { ki0 = 0; ++tap; }
  }

  const int mhi = (lane >> 4) << 3;
#pragma unroll
  for (int t = 0; t < 8; ++t) {
    const int col = t * 16 + nIdx;
    const float bv = bias[col];
#pragma unroll
    for (int v = 0; v < 8; ++v) {
      const float val = fmaxf(acc[t][v] + bv, 0.0f);
      out[(size_t)(rowBase + mhi + v) * 128 + col] = val;
    }
  }
}

// ---------------------------------------------------------------------------
// Heads (tiny FLOPs): Dense K->2 softmax, K->6 tanh, K->16 softmax
// ---------------------------------------------------------------------------
__global__ void flags_head(const float* __restrict__ h, const float* __restrict__ w,
                           const float* __restrict__ b, float* __restrict__ out,
                           int N) {
  const int n = blockIdx.x * blockDim.x + threadIdx.x;
  if (n >= N) return;
  const float* hr = h + (size_t)n * 128;
  float a0 = b[0], a1 = b[1];
  for (int k = 0; k < 128; ++k) {
    const float hv = hr[k];
    a0 += hv * w[k * 2 + 0];
    a1 += hv * w[k * 2 + 1];
  }
  const float m  = fmaxf(a0, a1);
  const float e0 = __expf(a0 - m), e1 = __expf(a1 - m);
  const float inv = 1.0f / (e0 + e1);
  out[(size_t)n * 2 + 0] = e0 * inv;
  out[(size_t)n * 2 + 1] = e1 * inv;
}

template <int F, bool SOFTMAX>
__global__ void dense_head(const float* __restrict__ h, const float* __restrict__ w,
                           const float* __restrict__ b, float* __restrict__ out,
                           int M) {
  const int n = blockIdx.x * blockDim.x + threadIdx.x;
  if (n >= M) return;
  const float* hr = h + (size_t)n * 128;
  float acc[F];
#pragma unroll
  for (int f = 0; f < F; ++f) acc[f] = b[f];
  for (int k = 0; k < 128; ++k) {
    const float hv = hr[k];
#pragma unroll
    for (int f = 0; f < F; ++f) acc[f] += hv * w[k * F + f];
  }
  if (SOFTMAX) {
    float m = acc[0];
#pragma unroll
    for (int f = 1; f < F; ++f) m = fmaxf(m, acc[f]);
    float s = 0.0f;
#pragma unroll
    for (int f = 0; f < F; ++f) { acc[f] = __expf(acc[f] - m); s += acc[f]; }
    const float inv = 1.0f / s;
#pragma unroll
    for (int f = 0; f < F; ++f) out[(size_t)n * F + f] = acc[f] * inv;
  } else {
#pragma unroll
    for (int f = 0; f < F; ++f) out[(size_t)n * F + f] = tanhf(acc[f]);
  }
}

// ---------------------------------------------------------------------------
extern "C" void kernel_launch(void* const* d_in, const int* in_sizes, int n_in,
                              void* d_out, int out_size, void* d_ws, size_t ws_size,
                              hipStream_t stream) {
  const float* x        = (const float*)d_in[0];
  const float* pre_w0   = (const float*)d_in[1];
  const float* pre_b0   = (const float*)d_in[2];
  const float* pre_w1   = (const float*)d_in[3];
  const float* pre_b1   = (const float*)d_in[4];
  const float* merge_w0 = (const float*)d_in[5];
  const float* merge_b0 = (const float*)d_in[6];
  const float* post_w0  = (const float*)d_in[7];
  const float* post_b0  = (const float*)d_in[8];
  const float* dec_w0   = (const float*)d_in[9];
  const float* dec_b0   = (const float*)d_in[10];
  const float* merge_w1 = (const float*)d_in[11];
  const float* merge_b1 = (const float*)d_in[12];
  const float* post_w1  = (const float*)d_in[13];
  const float* post_b1  = (const float*)d_in[14];
  const float* dec_w1   = (const float*)d_in[15];
  const float* dec_b1   = (const float*)d_in[16];
  const float* flags_w  = (const float*)d_in[17];
  const float* flags_b  = (const float*)d_in[18];
  const float* mmerge_w = (const float*)d_in[19];
  const float* mmerge_b = (const float*)d_in[20];
  const float* mean_w   = (const float*)d_in[21];
  const float* mean_b   = (const float*)d_in[22];
  const float* smerge_w = (const float*)d_in[23];
  const float* smerge_b = (const float*)d_in[24];
  const float* sym_w    = (const float*)d_in[25];
  const float* sym_b    = (const float*)d_in[26];

  const int N  = in_sizes[0] / 72;   // 262144
  const int M4 = N / 4;              // 65536

  // Workspace layout (fp32): two [N,128] + two [N/4,128] buffers (~336 MB)
  float* ws = (float*)d_ws;
  const size_t NB = (size_t)N * 128;
  const size_t MB = (size_t)M4 * 128;
  float* W0 = ws;
  float* W1 = ws + NB;
  float* W2 = ws + 2 * NB;
  float* W3 = ws + 2 * NB + MB;

  const dim3 blk(256);
  // dynamic LDS: full weight slab, float2-paired  (R*128 floats)
  const size_t sh72  = (size_t)(3 * 72)  * 128 * 4;  // 108 KB
  const size_t sh128 = (size_t)(3 * 128) * 128 * 4;  // 192 KB
  const size_t sh512 = (size_t)512 * 128 * 4;        // 256 KB
  const size_t shDec = (size_t)128 * 128 * 4;        // 64 KB

  // trunk
  conv3_relu_wmma<<<dim3(N / 128), blk, sh72, stream>>>(x, pre_w0, pre_b0, W0, N, 72);
  conv3_relu_wmma<<<dim3(N / 128), blk, sh128, stream>>>(W0, pre_w1, pre_b1, W1, N, 128);

  gemm_relu_wmma<0><<<dim3(M4 / 128, 1), blk, sh512, stream>>>(W1, merge_w0, merge_b0, W2, M4, 512, 128);
  conv3_relu_wmma<<<dim3(M4 / 128), blk, sh128, stream>>>(W2, post_w0, post_b0, W3, M4, 128);
  gemm_relu_wmma<1><<<dim3(M4 / 128, 4), blk, shDec, stream>>>(W3, dec_w0, dec_b0, W0, M4, 128, 512);

  gemm_relu_wmma<0><<<dim3(M4 / 128, 1), blk, sh512, stream>>>(W0, merge_w1, merge_b1, W2, M4, 512, 128);
  conv3_relu_wmma<<<dim3(M4 / 128), blk, sh128, stream>>>(W2, post_w1, post_b1, W3, M4, 128);
  gemm_relu_wmma<1><<<dim3(M4 / 128, 4), blk, shDec, stream>>>(W3, dec_w1, dec_b1, W1, M4, 128, 512);
  // h (trunk output) lives in W1

  // outputs: flags [N*2] | symbols [M4*16] | means [M4*6]
  float* out       = (float*)d_out;
  float* out_flags = out;
  float* out_syms  = out + (size_t)N * 2;
  float* out_means = out + (size_t)N * 2 + (size_t)M4 * 16;

  flags_head<<<dim3((N + 255) / 256), blk, 0, stream>>>(W1, flags_w, flags_b, out_flags, N);

  gemm_relu_wmma<0><<<dim3(M4 / 128, 1), blk, sh512, stream>>>(W1, mmerge_w, mmerge_b, W2, M4, 512, 128);
  dense_head<6, false><<<dim3((M4 + 255) / 256), blk, 0, stream>>>(W2, mean_w, mean_b, out_means, M4);

  gemm_relu_wmma<0><<<dim3(M4 / 128, 1), blk, sh512, stream>>>(W1, smerge_w, smerge_b, W3, M4, 512, 128);
  dense_head<16, true><<<dim3((M4 + 255) / 256), blk, 0, stream>>>(W3, sym_w, sym_b, out_syms, M4);
}